// YOLOv3Loss_19430432047615
// MI455X (gfx1250) — compile-verified
//
#include <hip/hip_runtime.h>
#include <math.h>

// ---------------- problem constants ----------------
#define BN   32
#define AN   3
#define GN   50
#define HN   52
#define WN   52
#define CN   80
#define HW   (HN * WN)            // 2704
#define NCELL (BN * AN * HW)      // 259584 = 1014 * 256 exactly
#define NBLK  (NCELL / 256)       // 1014

typedef float v2f __attribute__((ext_vector_type(2)));
typedef float v8f __attribute__((ext_vector_type(8)));

// ---------------- WMMA-based wave32 sum ----------------
// D = A(16x4) x B(4x16) + 0 with B = all ones  =>  D[m,n] = rowsum(A[m,:]).
// The 64 A slots (2 VGPRs x 32 lanes) biject onto the 16x4 matrix, so
// sum over all 256 D entries == 16 * sum over all lanes of (a0 + a1),
// independent of the exact lane<->element mapping. We put v in a0, 0 in a1,
// per-lane-sum the 8 D registers, butterfly, and scale by 1/16. Exact scale.
__device__ __forceinline__ float wave_reduce_sum(float v) {
  v2f a; a[0] = v;    a[1] = 0.0f;
  v2f b; b[0] = 1.0f; b[1] = 1.0f;
  v8f c = {};
  v8f d = __builtin_amdgcn_wmma_f32_16x16x4_f32(
      /*neg_a=*/false, a, /*neg_b=*/false, b,
      /*c_mod=*/(short)0, c, /*reuse_a=*/false, /*reuse_b=*/false);
  float s = d[0] + d[1] + d[2] + d[3] + d[4] + d[5] + d[6] + d[7];
  #pragma unroll
  for (int off = 16; off >= 1; off >>= 1) s += __shfl_xor(s, off, 32);
  return s * 0.0625f; // /16, exact
}

__device__ __forceinline__ float sigmoidf(float x) {
  return 1.0f / (1.0f + __expf(-x));
}

// torch BCELoss with log clamped at -100 (matches reference _bce)
__device__ __forceinline__ float bcef(float p, float t) {
  float lp = fmaxf(logf(p), -100.0f);
  float lq = fmaxf(logf(1.0f - p), -100.0f);
  return -(t * lp + (1.0f - t) * lq);
}

__device__ __forceinline__ int cell_idx(int b, int a, int j, int i) {
  return ((b * AN + a) * HN + j) * WN + i;
}

// ---------------- K0: init target arrays ----------------
__global__ void yolo_init(float* __restrict__ mask, float* __restrict__ noobj,
                          float* __restrict__ tx, float* __restrict__ ty,
                          float* __restrict__ tw, float* __restrict__ th,
                          unsigned* __restrict__ cbits) {
  int idx = blockIdx.x * 256 + threadIdx.x;
  if (idx < NCELL) {
    mask[idx] = 0.0f; noobj[idx] = 1.0f;
    tx[idx] = 0.0f; ty[idx] = 0.0f; tw[idx] = 0.0f; th[idx] = 0.0f;
    cbits[3 * idx + 0] = 0u; cbits[3 * idx + 1] = 0u; cbits[3 * idx + 2] = 0u;
  }
}

// ---------------- K1: build targets (lane b owns image b; sequential gts) ----
__global__ void yolo_build(const float* __restrict__ tg,
                           float* __restrict__ mask, float* __restrict__ noobj,
                           float* __restrict__ tx, float* __restrict__ ty,
                           float* __restrict__ tw, float* __restrict__ th,
                           unsigned* __restrict__ cbits) {
  int b = threadIdx.x;
  if (b >= BN) return;
  const float sax[3] = {1.25f, 2.0f, 4.125f};   // ANCHORS / stride(8)
  const float say[3] = {1.625f, 3.75f, 2.875f};
  for (int g = 0; g < GN; ++g) {
    const float* t = tg + (b * GN + g) * 5;
    float cls = t[0];
    // replicate reference rounding: (v / 416) * 52
    float gx = (t[1] / 416.0f) * 52.0f;
    float gy = (t[2] / 416.0f) * 52.0f;
    float gw = (t[3] / 416.0f) * 52.0f;
    float gh = (t[4] / 416.0f) * 52.0f;
    int gi = (int)gx; gi = gi < 0 ? 0 : (gi > WN - 1 ? WN - 1 : gi);
    int gj = (int)gy; gj = gj < 0 ? 0 : (gj > HN - 1 ? HN - 1 : gj);
    float area_g = (gw + 1.0f) * (gh + 1.0f);
    int best = 0; float best_iou = -1.0f;
    #pragma unroll
    for (int a = 0; a < AN; ++a) {
      float iw = fmaxf(fminf(gw, sax[a]) + 1.0f, 0.0f);
      float ih = fmaxf(fminf(gh, say[a]) + 1.0f, 0.0f);
      float inter = iw * ih;
      float area_a = (sax[a] + 1.0f) * (say[a] + 1.0f);
      float iou = inter / (area_g + area_a - inter + 1e-16f);
      if (iou > best_iou) { best_iou = iou; best = a; }           // first-max wins
      if (iou > 0.5f) noobj[cell_idx(b, a, gj, gi)] = 0.0f;       // min-scatter
    }
    int cc = cell_idx(b, best, gj, gi);
    mask[cc] = 1.0f;
    tx[cc] = gx - (float)gi;
    ty[cc] = gy - (float)gj;
    tw[cc] = logf(gw / sax[best] + 1e-16f);
    th[cc] = logf(gh / say[best] + 1e-16f);
    int ci = (int)cls; ci = ci < 0 ? 0 : (ci > CN - 1 ? CN - 1 : ci);
    cbits[cc * 3 + (ci >> 5)] |= (1u << (ci & 31));               // tcls set-scatter
  }
}

// ---------------- K2: main streaming loss reduction ----------------
// partial sums q: 0=x 1=y 2=w 3=h 4=conf_obj 5=conf_noobj 6=npos 7=cls
__global__ void yolo_main(const float* __restrict__ in,
                          const float* __restrict__ mask, const float* __restrict__ noobj,
                          const float* __restrict__ tx, const float* __restrict__ ty,
                          const float* __restrict__ tw, const float* __restrict__ th,
                          const unsigned* __restrict__ cbits,
                          float* __restrict__ partials) {
  int cell = blockIdx.x * 256 + threadIdx.x;   // NCELL divisible by 256: no tail
  int i = cell % WN; int r = cell / WN;
  int j = r % HN;    r /= HN;
  int a = r % AN;    int b = r / AN;

  // inputs.reshape(B, A, 85, H, W): attr stride = HW
  const float* base = in + (size_t)(b * 255 + a * 85) * HW + j * WN + i;

  float px = sigmoidf(base[0]);
  float py = sigmoidf(base[1 * HW]);
  float pw = base[2 * HW];
  float ph = base[3 * HW];
  float pc = sigmoidf(base[4 * HW]);

  float m  = mask[cell];
  float no = noobj[cell];

  float s0 = bcef(px * m, tx[cell] * m);
  float s1 = bcef(py * m, ty[cell] * m);
  float dw = pw * m - tw[cell] * m; float s2 = dw * dw;
  float dh = ph * m - th[cell] * m; float s3 = dh * dh;
  float s4 = bcef(pc * m, m);
  float s5 = bcef(pc * no, 0.0f);
  float s6 = m;
  float s7 = 0.0f;
  if (m != 0.0f) {  // class logits only touched at masked cells (~1600 of 259584)
    unsigned w0 = cbits[cell * 3 + 0];
    unsigned w1 = cbits[cell * 3 + 1];
    unsigned w2 = cbits[cell * 3 + 2];
    #pragma unroll 4
    for (int c = 0; c < CN; ++c) {
      float p = sigmoidf(base[(5 + c) * HW]);
      unsigned word = (c < 32) ? w0 : (c < 64 ? w1 : w2);
      float t = (float)((word >> (c & 31)) & 1u);
      s7 += bcef(p, t);
    }
  }

  // wave reduction via WMMA (EXEC all ones here: reconverged, no tail)
  float acc[8] = {s0, s1, s2, s3, s4, s5, s6, s7};
  #pragma unroll
  for (int q = 0; q < 8; ++q) acc[q] = wave_reduce_sum(acc[q]);

  __shared__ float red[8][8];
  int wave = threadIdx.x >> 5, lane = threadIdx.x & 31;
  if (lane == 0) {
    #pragma unroll
    for (int q = 0; q < 8; ++q) red[wave][q] = acc[q];
  }
  __syncthreads();
  if (threadIdx.x < 8) {
    float t = 0.0f;
    #pragma unroll
    for (int w = 0; w < 8; ++w) t += red[w][threadIdx.x];  // fixed order: deterministic
    partials[(size_t)blockIdx.x * 8 + threadIdx.x] = t;
  }
}

// ---------------- K3: deterministic final reduction + loss formula ----------
__global__ void yolo_final(const float* __restrict__ partials, float* __restrict__ out) {
  float acc[8] = {0, 0, 0, 0, 0, 0, 0, 0};
  for (int idx = threadIdx.x; idx < NBLK; idx += 256) {
    #pragma unroll
    for (int q = 0; q < 8; ++q) acc[q] += partials[(size_t)idx * 8 + q];
  }
  #pragma unroll
  for (int q = 0; q < 8; ++q) acc[q] = wave_reduce_sum(acc[q]);  // WMMA path

  __shared__ float red[8][8];
  int wave = threadIdx.x >> 5, lane = threadIdx.x & 31;
  if (lane == 0) {
    #pragma unroll
    for (int q = 0; q < 8; ++q) red[wave][q] = acc[q];
  }
  __syncthreads();
  if (threadIdx.x == 0) {
    float s[8];
    #pragma unroll
    for (int q = 0; q < 8; ++q) {
      float t = 0.0f;
      #pragma unroll
      for (int w = 0; w < 8; ++w) t += red[w][q];
      s[q] = t;
    }
    const float Nf = (float)NCELL;
    float lx = s[0] / Nf, ly = s[1] / Nf;
    float lw = s[2] / Nf, lh = s[3] / Nf;
    float lconf = s[4] / Nf + 0.5f * (s[5] / Nf);
    float lcls = s[7] / fmaxf(s[6] * (float)CN, 1.0f);
    out[0] = 2.5f * (lx + ly) + 2.5f * (lw + lh) + lconf + lcls;
  }
}

// ---------------- host launch ----------------
extern "C" void kernel_launch(void* const* d_in, const int* in_sizes, int n_in,
                              void* d_out, int out_size, void* d_ws, size_t ws_size,
                              hipStream_t stream) {
  (void)in_sizes; (void)n_in; (void)out_size; (void)ws_size;
  const float* inputs  = (const float*)d_in[0];   // [32,255,52,52]
  const float* targets = (const float*)d_in[1];   // [32,50,5]

  float* wsf   = (float*)d_ws;
  float* mask  = wsf + (size_t)0 * NCELL;
  float* noobj = wsf + (size_t)1 * NCELL;
  float* tx    = wsf + (size_t)2 * NCELL;
  float* ty    = wsf + (size_t)3 * NCELL;
  float* tw    = wsf + (size_t)4 * NCELL;
  float* th    = wsf + (size_t)5 * NCELL;
  unsigned* cbits = (unsigned*)(wsf + (size_t)6 * NCELL);      // 3*NCELL u32
  float* partials = wsf + (size_t)9 * NCELL;                   // NBLK*8 floats

  yolo_init <<<NBLK, 256, 0, stream>>>(mask, noobj, tx, ty, tw, th, cbits);
  yolo_build<<<1,    32, 0, stream>>>(targets, mask, noobj, tx, ty, tw, th, cbits);
  yolo_main <<<NBLK, 256, 0, stream>>>(inputs, mask, noobj, tx, ty, tw, th, cbits, partials);
  yolo_final<<<1,   256, 0, stream>>>(partials, (float*)d_out);
}